// Net_33560874451268
// MI455X (gfx1250) — compile-verified
//
#include <hip/hip_runtime.h>
#include <hip/hip_bf16.h>
#include <math.h>

typedef __attribute__((ext_vector_type(2))) float v2f;
typedef __attribute__((ext_vector_type(8))) float v8f;

#define N_NODES 100000
#define N_EDGES 1600000
#define IN_F    67
#define H1C     10
#define C1C     8
#define F1      80            // H1*C1
#define E_TOT   (N_EDGES + N_NODES)
#define ENC_NEG_INF 0x007FFFFFu   // enc(-inf)

// ---- order-preserving float<->uint map for atomic max ----
static __device__ __forceinline__ unsigned enc_f32(float x) {
    unsigned u = __float_as_uint(x);
    return (u & 0x80000000u) ? ~u : (u | 0x80000000u);
}
static __device__ __forceinline__ float dec_f32(unsigned u) {
    return __uint_as_float((u & 0x80000000u) ? (u ^ 0x80000000u) : ~u);
}
static __device__ __forceinline__ float lrelu(float v) { return v > 0.f ? v : 0.2f * v; }

// ---------------- fill ----------------
__global__ __launch_bounds__(256) void fill_u32(unsigned* __restrict__ p, unsigned v, int n) {
    int i = blockIdx.x * blockDim.x + threadIdx.x;
    if (i < n) p[i] = v;
}

// ---------------- layer-1 GEMM: h1[N,80] = x[N,67] @ W1[67,80] via f32 WMMA ----------------
__global__ __launch_bounds__(256) void gemm1_wmma(const float* __restrict__ x,
                                                  const float* __restrict__ W,
                                                  float* __restrict__ h) {
    const int lane = threadIdx.x & 31;
    const int wave = blockIdx.x * (blockDim.x >> 5) + (threadIdx.x >> 5);
    const int MT = N_NODES / 16;   // 6250 (exact)
    const int NT = F1 / 16;        // 5 (exact)
    if (wave >= MT * NT) return;   // whole-wave exit: EXEC stays all-ones for WMMA
    const int mt = wave / NT, nt = wave % NT;
    const int row  = mt * 16 + (lane & 15);          // A rows
    const int col  = nt * 16 + (lane & 15);          // B cols
    const int ksel = (lane >> 4) ? 2 : 0;            // lanes 16-31 hold K+2,K+3
    v8f c = {};
    // full K-steps: K = 0..63 (no bounds checks)
    for (int k0 = 0; k0 < 64; k0 += 4) {
        const int ka = k0 + ksel;
        v2f a, b;
        a.x = x[row * IN_F + ka];
        a.y = x[row * IN_F + ka + 1];
        b.x = W[ka * F1 + col];
        b.y = W[(ka + 1) * F1 + col];
        c = __builtin_amdgcn_wmma_f32_16x16x4_f32(false, a, false, b, (short)0, c, false, false);
    }
    { // tail K-step: K = 64..66 valid, 67 zero-padded
        const int ka = 64 + ksel;
        v2f a, b;
        a.x = (ka     < IN_F) ? x[row * IN_F + ka]     : 0.f;
        a.y = (ka + 1 < IN_F) ? x[row * IN_F + ka + 1] : 0.f;
        b.x = (ka     < IN_F) ? W[ka * F1 + col]       : 0.f;
        b.y = (ka + 1 < IN_F) ? W[(ka + 1) * F1 + col] : 0.f;
        c = __builtin_amdgcn_wmma_f32_16x16x4_f32(false, a, false, b, (short)0, c, false, false);
    }
    // C/D layout: VGPR v -> row v (lanes 0-15) / row v+8 (lanes 16-31)
    const int rbase = mt * 16 + ((lane >> 4) ? 8 : 0);
    #pragma unroll
    for (int v = 0; v < 8; ++v)
        h[(rbase + v) * F1 + col] = c[v];
}

// ---------------- per-node attention coefficients, layer 1 ----------------
__global__ __launch_bounds__(256) void attn_coef1(const float* __restrict__ h,
                                                  const float* __restrict__ att_src,
                                                  const float* __restrict__ att_dst,
                                                  float* __restrict__ a_src,
                                                  float* __restrict__ a_dst) {
    int n = blockIdx.x * blockDim.x + threadIdx.x;
    if (n >= N_NODES) return;
    const float* hr = h + (long long)n * F1;
    #pragma unroll
    for (int hh = 0; hh < H1C; ++hh) {
        float ss = 0.f, sd = 0.f;
        #pragma unroll
        for (int cc = 0; cc < C1C; ++cc) {
            float v = hr[hh * C1C + cc];
            ss += v * att_src[hh * C1C + cc];
            sd += v * att_dst[hh * C1C + cc];
        }
        a_src[n * H1C + hh] = ss;
        a_dst[n * H1C + hh] = sd;
    }
}

// ---------------- edge helpers ----------------
template <int H>
static __device__ __forceinline__ void edge_decode(int t, int& e, int& hh, int& s, int& d,
                                                   const int* __restrict__ ei) {
    e = t / H; hh = t - e * H;
    if (e < N_EDGES) { s = ei[e]; d = ei[N_EDGES + e]; }
    else             { s = d = e - N_EDGES; }          // virtual self-loops
}

template <int H>
__global__ __launch_bounds__(256) void edge_max(const int* __restrict__ ei,
                                                const float* __restrict__ a_src,
                                                const float* __restrict__ a_dst,
                                                unsigned* __restrict__ m) {
    int t = blockIdx.x * blockDim.x + threadIdx.x;
    if (t >= E_TOT * H) return;
    int e, hh, s, d; edge_decode<H>(t, e, hh, s, d, ei);
    float v = lrelu(a_src[s * H + hh] + a_dst[d * H + hh]);
    atomicMax(&m[d * H + hh], enc_f32(v));
}

template <int H>
__global__ __launch_bounds__(256) void edge_sum(const int* __restrict__ ei,
                                                const float* __restrict__ a_src,
                                                const float* __restrict__ a_dst,
                                                const unsigned* __restrict__ m,
                                                float* __restrict__ den) {
    int t = blockIdx.x * blockDim.x + threadIdx.x;
    if (t >= E_TOT * H) return;
    int e, hh, s, d; edge_decode<H>(t, e, hh, s, d, ei);
    float v  = lrelu(a_src[s * H + hh] + a_dst[d * H + hh]);
    float ex = expf(v - dec_f32(m[d * H + hh]));
    atomicAdd(&den[d * H + hh], ex);
}

template <int H, int C>
__global__ __launch_bounds__(256) void edge_scatter(const int* __restrict__ ei,
                                                    const float* __restrict__ a_src,
                                                    const float* __restrict__ a_dst,
                                                    const unsigned* __restrict__ m,
                                                    const float* __restrict__ den,
                                                    const float* __restrict__ hsrc,
                                                    float* __restrict__ out) {
    int t = blockIdx.x * blockDim.x + threadIdx.x;
    if (t >= E_TOT * H) return;
    int e, hh, s, d; edge_decode<H>(t, e, hh, s, d, ei);
    float v     = lrelu(a_src[s * H + hh] + a_dst[d * H + hh]);
    float alpha = expf(v - dec_f32(m[d * H + hh])) / den[d * H + hh];
    const float* hr = hsrc + ((long long)s * H + hh) * C;
    float*       orow = out + ((long long)d * H + hh) * C;
    #pragma unroll
    for (int cc = 0; cc < C; ++cc)
        atomicAdd(&orow[cc], hr[cc] * alpha);
}

// ---------------- ELU(o1 + b1) in place ----------------
__global__ __launch_bounds__(256) void elu_bias(float* __restrict__ o, const float* __restrict__ b) {
    int i = blockIdx.x * blockDim.x + threadIdx.x;
    if (i >= N_NODES * F1) return;
    float v = o[i] + b[i % F1];
    o[i] = v > 0.f ? v : expm1f(v);
}

// ---------------- layer-2 matvec + attention coefficients (H=1,C=1) ----------------
__global__ __launch_bounds__(256) void gemm2(const float* __restrict__ hm, const float* __restrict__ W2,
                                             const float* __restrict__ att_src2,
                                             const float* __restrict__ att_dst2,
                                             float* __restrict__ h2,
                                             float* __restrict__ as2, float* __restrict__ ad2) {
    int n = blockIdx.x * blockDim.x + threadIdx.x;
    if (n >= N_NODES) return;
    const float* hr = hm + (long long)n * F1;
    float sacc = 0.f;
    #pragma unroll
    for (int f = 0; f < F1; ++f) sacc += hr[f] * W2[f];
    h2[n]  = sacc;
    as2[n] = sacc * att_src2[0];
    ad2[n] = sacc * att_dst2[0];
}

__global__ __launch_bounds__(256) void finalize(const float* __restrict__ o2,
                                                const float* __restrict__ b2,
                                                float* __restrict__ out) {
    int n = blockIdx.x * blockDim.x + threadIdx.x;
    if (n < N_NODES) out[n] = o2[n] + b2[0];
}

static inline int cdiv(long long a, long long b) { return (int)((a + b - 1) / b); }

extern "C" void kernel_launch(void* const* d_in, const int* in_sizes, int n_in,
                              void* d_out, int out_size, void* d_ws, size_t ws_size,
                              hipStream_t stream) {
    (void)in_sizes; (void)n_in; (void)out_size; (void)ws_size;
    const float* x        = (const float*)d_in[0];
    const int*   ei       = (const int*)  d_in[1];   // [2, E]
    const float* W1       = (const float*)d_in[2];
    const float* att_src1 = (const float*)d_in[3];
    const float* att_dst1 = (const float*)d_in[4];
    const float* b1       = (const float*)d_in[5];
    const float* W2       = (const float*)d_in[6];
    const float* att_src2 = (const float*)d_in[7];
    const float* att_dst2 = (const float*)d_in[8];
    const float* b2       = (const float*)d_in[9];

    float* ws = (float*)d_ws;                       // ~82.4 MB total
    float*    h1  = ws;                              // 8,000,000
    float*    o1  = ws + 8000000;                    // 8,000,000 (becomes ELU output)
    float*    as1 = ws + 16000000;                   // 1,000,000
    float*    ad1 = ws + 17000000;                   // 1,000,000
    unsigned* m1  = (unsigned*)(ws + 18000000);      // 1,000,000
    float*    dn1 = ws + 19000000;                   // 1,000,000
    float*    h2  = ws + 20000000;                   // 100,000
    float*    as2 = ws + 20100000;
    float*    ad2 = ws + 20200000;
    unsigned* m2  = (unsigned*)(ws + 20300000);
    float*    dn2 = ws + 20400000;
    float*    o2  = ws + 20500000;

    const int BT = 256;

    // init accumulators (fresh every call)
    fill_u32<<<cdiv(8000000, BT), BT, 0, stream>>>((unsigned*)o1, 0u, 8000000);
    fill_u32<<<cdiv(1000000, BT), BT, 0, stream>>>(m1, ENC_NEG_INF, 1000000);
    fill_u32<<<cdiv(1000000, BT), BT, 0, stream>>>((unsigned*)dn1, 0u, 1000000);
    fill_u32<<<cdiv(100000,  BT), BT, 0, stream>>>(m2, ENC_NEG_INF, 100000);
    fill_u32<<<cdiv(100000,  BT), BT, 0, stream>>>((unsigned*)dn2, 0u, 100000);
    fill_u32<<<cdiv(100000,  BT), BT, 0, stream>>>((unsigned*)o2, 0u, 100000);

    // layer 1
    const int waves1 = (N_NODES / 16) * (F1 / 16);   // 31250
    gemm1_wmma<<<cdiv(waves1, BT / 32), BT, 0, stream>>>(x, W1, h1);
    attn_coef1<<<cdiv(N_NODES, BT), BT, 0, stream>>>(h1, att_src1, att_dst1, as1, ad1);
    edge_max<H1C>    <<<cdiv((long long)E_TOT * H1C, BT), BT, 0, stream>>>(ei, as1, ad1, m1);
    edge_sum<H1C>    <<<cdiv((long long)E_TOT * H1C, BT), BT, 0, stream>>>(ei, as1, ad1, m1, dn1);
    edge_scatter<H1C, C1C><<<cdiv((long long)E_TOT * H1C, BT), BT, 0, stream>>>(ei, as1, ad1, m1, dn1, h1, o1);
    elu_bias<<<cdiv((long long)N_NODES * F1, BT), BT, 0, stream>>>(o1, b1);

    // layer 2 (H=1, C=1)
    gemm2<<<cdiv(N_NODES, BT), BT, 0, stream>>>(o1, W2, att_src2, att_dst2, h2, as2, ad2);
    edge_max<1>    <<<cdiv(E_TOT, BT), BT, 0, stream>>>(ei, as2, ad2, m2);
    edge_sum<1>    <<<cdiv(E_TOT, BT), BT, 0, stream>>>(ei, as2, ad2, m2, dn2);
    edge_scatter<1, 1><<<cdiv(E_TOT, BT), BT, 0, stream>>>(ei, as2, ad2, m2, dn2, h2, o2);
    finalize<<<cdiv(N_NODES, BT), BT, 0, stream>>>(o2, b2, (float*)d_out);
}